// PointConv_11038065951507
// MI455X (gfx1250) — compile-verified
//
#include <hip/hip_runtime.h>
#include <hip/hip_bf16.h>

typedef __attribute__((ext_vector_type(2))) float v2f;
typedef __attribute__((ext_vector_type(8))) float v8f;

#define B_ 4
#define C_IN 128
#define C_OUT 256
#define N_ 4096

// ---------------------------------------------------------------------------
// Kernel 1: transpose x [B][C][N] -> xT [B][N][C] so gathered feature vectors
// are contiguous (512B) and WMMA B-fragments are single b64 loads per lane.
// ---------------------------------------------------------------------------
__global__ __launch_bounds__(256) void transpose_x(const float* __restrict__ x,
                                                   float* __restrict__ xT) {
    __shared__ float tile[32][33];
    int b  = blockIdx.z;
    int n0 = blockIdx.x * 32;   // N/32 = 128 blocks
    int c0 = blockIdx.y * 32;   // C/32 = 4 blocks
    for (int i = threadIdx.y; i < 32; i += 8)
        tile[i][threadIdx.x] = x[((size_t)b * C_IN + c0 + i) * N_ + n0 + threadIdx.x];
    __syncthreads();
    for (int i = threadIdx.y; i < 32; i += 8)
        xT[((size_t)b * N_ + n0 + i) * C_IN + c0 + threadIdx.x] = tile[threadIdx.x][i];
}

// ---------------------------------------------------------------------------
// Kernel 2: repack weight [256][128][1][9] into WMMA f32 16x16x4 A-fragment
// order: flat = ((kk*32 + c4)*16 + mt)*64 + lane*2 + j
//   lane<16 : (W[m, c0+0, kk], W[m, c0+1, kk])   m = mt*16 + (lane&15)
//   lane>=16: (W[m, c0+2, kk], W[m, c0+3, kk])   c0 = c4*4
// Only kk=0..7 (octant-7 column of the reference is provably always zero).
// ---------------------------------------------------------------------------
__global__ __launch_bounds__(256) void repack_weight(const float* __restrict__ w,
                                                     float* __restrict__ wrep) {
    int t = blockIdx.x * 256 + threadIdx.x;      // 0 .. 262143
    int j    = t & 1;
    int lane = (t >> 1) & 31;
    int mt   = (t >> 6) & 15;
    int c4   = (t >> 10) & 31;
    int kk   = t >> 15;                          // 0..7
    int m    = mt * 16 + (lane & 15);
    int ksub = ((lane >> 4) << 1) + j;           // 0..3
    int c    = c4 * 4 + ksub;
    wrep[t] = w[(size_t)m * (C_IN * 9) + c * 9 + kk];
}

// ---------------------------------------------------------------------------
// Kernel 3: ball query + per-octant first neighbor.
// gidx8[b][n][0]    = n (center, weight slice k=0)
// gidx8[b][n][1+o]  = global idx of first neighbor (among first 31 in-radius,
//                     index order) whose centered coords fall in octant o
//                     (o = 0..6), or -1. Octant 7 is always dead (center).
// ---------------------------------------------------------------------------
__global__ __launch_bounds__(256) void ball_octants(const float* __restrict__ pcs,
                                                    int* __restrict__ gidx8) {
#pragma clang fp contract(off)
    __shared__ float sx[N_], sy[N_], sz[N_];
    int b = blockIdx.y;
    const float* pb = pcs + (size_t)b * 3 * N_;
    for (int j = threadIdx.x; j < N_; j += 256) {
        sx[j] = pb[j];
        sy[j] = pb[N_ + j];
        sz[j] = pb[2 * N_ + j];
    }
    __syncthreads();

    int n = blockIdx.x * 256 + threadIdx.x;
    float xn = sx[n], yn = sy[n], zn = sz[n];
    float sqn = xn * xn + yn * yn + zn * zn;
    const float r2 = (float)(0.12 * 0.12);
    int f0 = -1, f1 = -1, f2 = -1, f3 = -1, f4 = -1, f5 = -1, f6 = -1;
    int cnt = 0, filled = 0;
    for (int j = 0; j < N_; ++j) {
        float xj = sx[j], yj = sy[j], zj = sz[j];
        float sqj = xj * xj + yj * yj + zj * zj;
        float dot = xn * xj + yn * yj + zn * zj;
        float d2  = (sqn + sqj) - 2.0f * dot;     // reference formula order
        if (d2 <= r2 && j != n) {
            ++cnt;
            float dx = xj - xn, dy = yj - yn, dz = zj - zn;
            int oct = ((dx >= 0.0f) ? 4 : 0) + ((dy >= 0.0f) ? 2 : 0) +
                      ((dz >= 0.0f) ? 1 : 0);
            int nb = ((1 << oct) & 0x7F) & ~filled;   // new first for octants 0..6
            f0 = (nb & 1)  ? j : f0;
            f1 = (nb & 2)  ? j : f1;
            f2 = (nb & 4)  ? j : f2;
            f3 = (nb & 8)  ? j : f3;
            f4 = (nb & 16) ? j : f4;
            f5 = (nb & 32) ? j : f5;
            f6 = (nb & 64) ? j : f6;
            filled |= nb;
            if (cnt >= 31 || filled == 0x7F) break;   // only first 31 slots count
        }
    }
    int* o = gidx8 + ((size_t)(b * N_ + n) << 3);
    o[0] = n;
    o[1] = f0; o[2] = f1; o[3] = f2; o[4] = f3; o[5] = f4; o[6] = f5; o[7] = f6;
}

// ---------------------------------------------------------------------------
// Kernel 4: WMMA GEMM. One wave handles one 16-point group x all 256 out
// channels (16 m-tiles of 16). D = sum_{kk=0..7} W_kk[256x128] * Xg_kk[128x16].
// A fragment (16x4 f32): lanes 0-15 = {K0,K1}, lanes 16-31 = {K2,K3}  (repacked)
// B fragment (4x16 f32): lanes 0-15 = rows {K0,K1} of col (lane&15),
//                        lanes 16-31 = rows {K2,K3}  -> one b64 from xT.
// ---------------------------------------------------------------------------
__global__ __launch_bounds__(128) void pointconv_gemm(
    const float* __restrict__ xT, const float* __restrict__ wrep,
    const float* __restrict__ bias, const int* __restrict__ gidx8,
    float* __restrict__ out) {
    int lane = threadIdx.x & 31;
    int wv   = threadIdx.x >> 5;
    int pg   = blockIdx.x * 4 + wv;          // 0..1023 point-groups
    int b    = pg >> 8;
    int n0   = (pg & 255) << 4;
    int pt   = lane & 15;
    int hi   = lane >> 4;                    // 0: rows K0,K1 / 1: rows K2,K3

    const int* gi = gidx8 + ((size_t)(b * N_ + n0 + pt) << 3);
    const float* xTb = xT + (size_t)b * N_ * C_IN;

    v8f acc[16];
#pragma unroll
    for (int mt = 0; mt < 16; ++mt)
        acc[mt] = (v8f){0.f, 0.f, 0.f, 0.f, 0.f, 0.f, 0.f, 0.f};

#pragma unroll
    for (int kk = 0; kk < 8; ++kk) {
        int g = gi[kk];
        bool valid = g >= 0;
        const float* col = xTb + (size_t)(valid ? g : 0) * C_IN + hi * 2;
        const float* ap  = wrep + (size_t)kk * 32768 + lane * 2;
        for (int c4 = 0; c4 < 32; ++c4) {
            v2f Bv = *(const v2f*)(col + c4 * 4);
            Bv.x = valid ? Bv.x : 0.0f;      // v_cndmask, no EXEC divergence
            Bv.y = valid ? Bv.y : 0.0f;
#pragma unroll
            for (int mt = 0; mt < 16; ++mt) {
                v2f Av = *(const v2f*)(ap + mt * 64);
                acc[mt] = __builtin_amdgcn_wmma_f32_16x16x4_f32(
                    false, Av, false, Bv, (short)0, acc[mt], false, false);
            }
            ap += 1024;
        }
    }

    // Epilogue: D layout (16x16 f32): VGPR r, lanes<16 -> M=r, lanes>=16 -> M=r+8
    int mBase = hi * 8;
    float* op = out + (size_t)b * C_OUT * N_ + n0 + pt;
#pragma unroll
    for (int mt = 0; mt < 16; ++mt) {
#pragma unroll
        for (int r = 0; r < 8; ++r) {
            int m = mt * 16 + mBase + r;
            op[(size_t)m * N_] = acc[mt][r] + bias[m];
        }
    }
}

// ---------------------------------------------------------------------------
extern "C" void kernel_launch(void* const* d_in, const int* in_sizes, int n_in,
                              void* d_out, int out_size, void* d_ws, size_t ws_size,
                              hipStream_t stream) {
    const float* x    = (const float*)d_in[0];   // [4,128,4096]
    const float* pcs  = (const float*)d_in[1];   // [4,3,4096]
    const float* w    = (const float*)d_in[2];   // [256,128,1,9]
    const float* bias = (const float*)d_in[3];   // [256]
    float* out = (float*)d_out;                  // [4,256,4096]

    float* xT    = (float*)d_ws;                              // 8 MB
    float* wrep  = (float*)((char*)d_ws + (8u << 20));        // 1 MB
    int*   gidx8 = (int*)  ((char*)d_ws + (9u << 20));        // 0.5 MB

    transpose_x  <<<dim3(N_ / 32, C_IN / 32, B_), dim3(32, 8), 0, stream>>>(x, xT);
    repack_weight<<<1024, 256, 0, stream>>>(w, wrep);
    ball_octants <<<dim3(N_ / 256, B_), 256, 0, stream>>>(pcs, gidx8);
    pointconv_gemm<<<256, 128, 0, stream>>>(xT, wrep, bias, gidx8, out);
}